// BahdanauAttention_13761075216860
// MI455X (gfx1250) — compile-verified
//
#include <hip/hip_runtime.h>
#include <hip/hip_bf16.h>

// Problem constants (from reference): B=32, S=2048, H=1024, QS=1024, VS=2048
#define B_   32
#define S_   2048
#define H_   1024
#define QS_  1024
#define VS_  2048
#define SCHUNKS 16
#define SC_LEN  (S_ / SCHUNKS)   // 128

typedef __attribute__((ext_vector_type(2))) float v2f;
typedef __attribute__((ext_vector_type(4))) float f4;
typedef __attribute__((ext_vector_type(8))) float v8f;

__device__ __forceinline__ float fast_tanh(float x) {
#if __has_builtin(__builtin_amdgcn_tanhf)
  return __builtin_amdgcn_tanhf(x);   // gfx1250 v_tanh_f32 (1 trans op)
#else
  return tanhf(x);
#endif
}

// ---------------------------------------------------------------------------
// Kernel 1: q[B,H] = query[B,QS] @ Wq[QS,H] via V_WMMA_F32_16X16X4_F32.
// One wave per 16x16 tile of q. Tiles: (M=32)/16 x (H=1024)/16 = 2*64 = 128.
// A-frag layout (16x4 f32): lane = {M = lane&15}, VGPR0/1 hold K = 2*(lane>>4)
// and +1. B-frag (4x16) symmetric. C/D: VGPR r -> M = r + 8*(lane>>4).
// ---------------------------------------------------------------------------
__global__ __launch_bounds__(128) void qproj_wmma(const float* __restrict__ query,
                                                  const float* __restrict__ Wq,
                                                  float* __restrict__ qout) {
  const int wave = threadIdx.x >> 5;
  const int lane = threadIdx.x & 31;
  const int tile = blockIdx.x * 4 + wave;      // 0..127
  const int m0 = (tile & 1) * 16;              // 2 M-tiles
  const int n0 = (tile >> 1) * 16;             // 64 N-tiles
  const int rowA  = m0 + (lane & 15);
  const int colB  = n0 + (lane & 15);
  const int khalf = (lane >> 4) << 1;          // 0 for lanes 0-15, 2 for 16-31

  v8f c = {0.f, 0.f, 0.f, 0.f, 0.f, 0.f, 0.f, 0.f};
  for (int k0 = 0; k0 < QS_; k0 += 4) {
    const int kk = k0 + khalf;
    v2f a, b;
    a.x = query[rowA * QS_ + kk];
    a.y = query[rowA * QS_ + kk + 1];
    b.x = Wq[(size_t)kk * H_ + colB];
    b.y = Wq[(size_t)(kk + 1) * H_ + colB];
    // 8 args: (neg_a, A, neg_b, B, c_mod, C, reuse_a, reuse_b)
    c = __builtin_amdgcn_wmma_f32_16x16x4_f32(false, a, false, b,
                                              (short)0, c, false, false);
  }
  const int mBase = m0 + ((lane >> 4) << 3);
  const int n = n0 + (lane & 15);
#pragma unroll
  for (int r = 0; r < 8; ++r)
    qout[(mBase + r) * H_ + n] = c[r];
}

// ---------------------------------------------------------------------------
// Kernel 2: scores[b,s] = sum_h v_energy[h] * tanh(q[b,h] + proj_key[b,s,h]),
// masked to -inf. One wave per (b,s) row; streams 256 MB of proj_key with
// non-temporal float4 loads. Wave32 shuffle reduction.
// ---------------------------------------------------------------------------
__global__ __launch_bounds__(256) void energy_scores(const float* __restrict__ proj_key,
                                                     const int*   __restrict__ mask,
                                                     const float* __restrict__ qvec,
                                                     const float* __restrict__ v_energy,
                                                     float* __restrict__ scores) {
  const int wave = threadIdx.x >> 5;
  const int lane = threadIdx.x & 31;
  const int row  = blockIdx.x * 8 + wave;      // 0 .. B*S-1
  const int b    = row >> 11;                  // row / S_
  const float* pk = proj_key + (size_t)row * H_;
  const float* qb = qvec + (size_t)b * H_;

  float acc = 0.f;
#pragma unroll
  for (int i = 0; i < H_ / (32 * 4); ++i) {    // 8 iterations of float4
    const int h = (i * 32 + lane) * 4;
    f4 p = __builtin_nontemporal_load((const f4*)(pk + h));  // streaming
    f4 q = *(const f4*)(qb + h);                              // L2-resident
    f4 v = *(const f4*)(v_energy + h);                        // L2-resident
    acc += v.x * fast_tanh(q.x + p.x);
    acc += v.y * fast_tanh(q.y + p.y);
    acc += v.z * fast_tanh(q.z + p.z);
    acc += v.w * fast_tanh(q.w + p.w);
  }
#pragma unroll
  for (int off = 16; off > 0; off >>= 1)
    acc += __shfl_xor(acc, off, 32);
  if (lane == 0) {
    const int m = mask[row];                   // mask is [B,1,S] int32
    scores[row] = (m == 0) ? -__builtin_inff() : acc;
  }
}

// ---------------------------------------------------------------------------
// Kernel 3: row softmax over S=2048 per batch. Writes alphas directly into
// the alphas section of d_out. All-masked rows yield NaN, matching jax.
// ---------------------------------------------------------------------------
__global__ __launch_bounds__(256) void softmax_rows(const float* __restrict__ scores,
                                                    float* __restrict__ alphas) {
  __shared__ float red[256];
  const int b = blockIdx.x;
  const int t = threadIdx.x;
  const float* sc = scores + (size_t)b * S_;

  float mx = -__builtin_inff();
  for (int s = t; s < S_; s += 256) mx = fmaxf(mx, sc[s]);
  red[t] = mx; __syncthreads();
  for (int off = 128; off > 0; off >>= 1) {
    if (t < off) red[t] = fmaxf(red[t], red[t + off]);
    __syncthreads();
  }
  mx = red[0]; __syncthreads();

  float sum = 0.f;
  for (int s = t; s < S_; s += 256) sum += __expf(sc[s] - mx);
  red[t] = sum; __syncthreads();
  for (int off = 128; off > 0; off >>= 1) {
    if (t < off) red[t] += red[t + off];
    __syncthreads();
  }
  const float inv = 1.0f / red[0];

  for (int s = t; s < S_; s += 256)
    alphas[(size_t)b * S_ + s] = __expf(sc[s] - mx) * inv;
}

// ---------------------------------------------------------------------------
// Kernel 4a: partial context. context[b,v] = sum_s alphas[b,s]*value[b,s,v].
// S split into 16 chunks for parallelism (deterministic, no atomics).
// Block = (b, v-chunk of 1024, s-chunk of 128); thread owns 4 v via float4.
// Streams the 512 MB value tensor with coalesced non-temporal loads.
// ---------------------------------------------------------------------------
__global__ __launch_bounds__(256) void context_partial(const float* __restrict__ value,
                                                       const float* __restrict__ alphas,
                                                       float* __restrict__ partial) {
  const int sc = blockIdx.x & (SCHUNKS - 1);
  const int vc = (blockIdx.x >> 4) & 1;
  const int b  = blockIdx.x >> 5;
  const int t  = threadIdx.x;
  const int vbase = vc * 1024 + t * 4;

  const float* arow = alphas + (size_t)b * S_ + sc * SC_LEN;
  const float* vptr = value + ((size_t)b * S_ + (size_t)sc * SC_LEN) * VS_ + vbase;

  f4 acc = {0.f, 0.f, 0.f, 0.f};
  for (int s = 0; s < SC_LEN; ++s) {
    const float a = arow[s];                                    // scalar bcast
    f4 v = __builtin_nontemporal_load((const f4*)(vptr + (size_t)s * VS_));
    acc += a * v;
  }
  *(f4*)(partial + (((size_t)(b * 2 + vc) * SCHUNKS + sc) * 1024) + t * 4) = acc;
}

// ---------------------------------------------------------------------------
// Kernel 4b: reduce the 16 s-chunk partials into context (d_out front half).
// ---------------------------------------------------------------------------
__global__ __launch_bounds__(256) void context_reduce(const float* __restrict__ partial,
                                                      float* __restrict__ context) {
  const int idx = blockIdx.x * 256 + threadIdx.x;  // 0 .. B*VS-1
  const int b   = idx >> 11;                       // / VS_
  const int v   = idx & (VS_ - 1);
  const int vc  = v >> 10;
  const int vin = v & 1023;
  const float* p = partial + ((size_t)(b * 2 + vc) * SCHUNKS) * 1024 + vin;
  float s = 0.f;
#pragma unroll
  for (int c = 0; c < SCHUNKS; ++c) s += p[c * 1024];
  context[idx] = s;
}

// ---------------------------------------------------------------------------
extern "C" void kernel_launch(void* const* d_in, const int* in_sizes, int n_in,
                              void* d_out, int out_size, void* d_ws, size_t ws_size,
                              hipStream_t stream) {
  const float* query    = (const float*)d_in[0];   // [B,QS]
  const float* proj_key = (const float*)d_in[1];   // [B,S,H]
  const float* value    = (const float*)d_in[2];   // [B,S,VS]
  const int*   mask     = (const int*)  d_in[3];   // [B,1,S]
  const float* Wq       = (const float*)d_in[4];   // [QS,H]
  const float* v_energy = (const float*)d_in[5];   // [H]

  float* out     = (float*)d_out;
  float* context = out;                  // [B,1,VS] -> B*VS floats
  float* alphas  = out + B_ * VS_;       // [B,1,S]  -> B*S  floats

  // Workspace layout (≈4.6 MB): q | scores | partials
  float* qvec    = (float*)d_ws;                       // B*H
  float* scores  = qvec + B_ * H_;                     // B*S
  float* partial = scores + B_ * S_;                   // B*2*SCHUNKS*1024

  qproj_wmma     <<<32, 128, 0, stream>>>(query, Wq, qvec);
  energy_scores  <<<(B_ * S_) / 8, 256, 0, stream>>>(proj_key, mask, qvec,
                                                     v_energy, scores);
  softmax_rows   <<<B_, 256, 0, stream>>>(scores, alphas);
  context_partial<<<B_ * 2 * SCHUNKS, 256, 0, stream>>>(value, alphas, partial);
  context_reduce <<<(B_ * VS_) / 256, 256, 0, stream>>>(partial, context);
}